// SelfAttnMatch_50869592654061
// MI455X (gfx1250) — compile-verified
//
#include <hip/hip_runtime.h>

// SelfAttnMatch for B=8, L=2048, D=1024 (f32).
//   xp = relu(x @ W^T + b)              [proj_kernel,   WMMA f32 16x16x4]
//   S  = xp @ xp^T, diag=0, mask=-inf   [scores_kernel, WMMA f32 16x16x4]
//   S  = softmax(S, axis=-1)            [softmax_kernel]
//   out = S @ x                          [av_kernel,     WMMA f32 16x16x4]
// Workspace: xp = 64MB, S = 128MB (assumes ws_size >= 192MB).
//
// GEMM skeleton: 256 threads = 8 waves (4 in M x 2 in N), block tile
// 128x128, BK=32.  Each wave owns a 32x64 patch = 2x4 WMMA tiles
// (8 x v8f accumulators), giving 8 WMMAs per 6 LDS fragment loads per
// K-micro-step so the f32 XDL pipe, not LDS issue, is the limiter.

#define B_ 8
#define L_ 2048
#define D_ 1024

typedef __attribute__((ext_vector_type(2))) float v2f;
typedef __attribute__((ext_vector_type(8))) float v8f;

__device__ __forceinline__ v8f wmma4(v2f a, v2f b, v8f c) {
  return __builtin_amdgcn_wmma_f32_16x16x4_f32(
      false, a, false, b, (short)0, c, false, false);
}

// ---------------------------------------------------------------------------
// Kernel 1: xp[m][n] = relu( sum_k X[m][k] * W[n][k] + bias[n] )
//   M = B*L = 16384, N = D, K = D.  Grid (M/128, N/128).
// ---------------------------------------------------------------------------
__global__ __launch_bounds__(256)
void proj_kernel(const float* __restrict__ X, const float* __restrict__ W,
                 const float* __restrict__ bias, float* __restrict__ XP) {
  __shared__ float As[128 * 36];  // [m][k], padded stride 36
  __shared__ float Ns[128 * 36];  // [n][k] (B operand held as rows of W)

  const int tid  = threadIdx.x;
  const int lane = tid & 31, wid = tid >> 5;
  const int wm = wid & 3, wn = wid >> 2;     // 4x2 wave grid
  const int h = lane >> 4, lm = lane & 15;
  const int m0 = blockIdx.x * 128;
  const int n0 = blockIdx.y * 128;

  const int row = tid >> 1, col = (tid & 1) * 16;  // 128x32 tile: 16 floats/thread

  v8f acc[2][4];
  v8f z = {};
#pragma unroll
  for (int i = 0; i < 2; i++)
#pragma unroll
    for (int j = 0; j < 4; j++) acc[i][j] = z;

  for (int kb = 0; kb < D_; kb += 32) {
    const float4* gA = (const float4*)(X + (size_t)(m0 + row) * D_ + kb + col);
    float4* sA = (float4*)(&As[row * 36 + col]);
    sA[0] = gA[0]; sA[1] = gA[1]; sA[2] = gA[2]; sA[3] = gA[3];

    const float4* gN = (const float4*)(W + (size_t)(n0 + row) * D_ + kb + col);
    float4* sN = (float4*)(&Ns[row * 36 + col]);
    sN[0] = gN[0]; sN[1] = gN[1]; sN[2] = gN[2]; sN[3] = gN[3];
    __syncthreads();

#pragma unroll
    for (int kk = 0; kk < 32; kk += 4) {
      v2f a[2], b[4];
#pragma unroll
      for (int t = 0; t < 2; t++) {
        const int mr = (wm * 32 + t * 16 + lm) * 36 + kk + 2 * h;
        a[t][0] = As[mr]; a[t][1] = As[mr + 1];
      }
#pragma unroll
      for (int t = 0; t < 4; t++) {
        const int nr = (wn * 64 + t * 16 + lm) * 36 + kk + 2 * h;
        b[t][0] = Ns[nr]; b[t][1] = Ns[nr + 1];
      }
#pragma unroll
      for (int i = 0; i < 2; i++)
#pragma unroll
        for (int j = 0; j < 4; j++) acc[i][j] = wmma4(a[i], b[j], acc[i][j]);
    }
    __syncthreads();
  }

#pragma unroll
  for (int i = 0; i < 2; i++)
#pragma unroll
    for (int j = 0; j < 4; j++) {
      const int n = n0 + wn * 64 + j * 16 + lm;
      const float bv = bias[n];
#pragma unroll
      for (int r = 0; r < 8; r++) {
        const int m = m0 + wm * 32 + i * 16 + r + 8 * h;
        XP[(size_t)m * D_ + n] = fmaxf(acc[i][j][r] + bv, 0.0f);
      }
    }
}

// ---------------------------------------------------------------------------
// Kernel 2: S[b][i][j] = xp_b[i] . xp_b[j], diag -> 0, padded key j -> -inf
//   Per batch GEMM: M=N=L, K=D.  Grid (L/128, L/128, B).
// ---------------------------------------------------------------------------
__global__ __launch_bounds__(256)
void scores_kernel(const float* __restrict__ XP,
                   const unsigned char* __restrict__ mask,
                   float* __restrict__ S) {
  __shared__ float As[128 * 36];
  __shared__ float Ns[128 * 36];

  const int tid  = threadIdx.x;
  const int lane = tid & 31, wid = tid >> 5;
  const int wm = wid & 3, wn = wid >> 2;
  const int h = lane >> 4, lm = lane & 15;
  const int m0 = blockIdx.x * 128;
  const int n0 = blockIdx.y * 128;
  const int bz = blockIdx.z;
  const float* Xb = XP + (size_t)bz * L_ * D_;

  const int row = tid >> 1, col = (tid & 1) * 16;

  v8f acc[2][4];
  v8f z = {};
#pragma unroll
  for (int i = 0; i < 2; i++)
#pragma unroll
    for (int j = 0; j < 4; j++) acc[i][j] = z;

  for (int kb = 0; kb < D_; kb += 32) {
    const float4* gA = (const float4*)(Xb + (size_t)(m0 + row) * D_ + kb + col);
    float4* sA = (float4*)(&As[row * 36 + col]);
    sA[0] = gA[0]; sA[1] = gA[1]; sA[2] = gA[2]; sA[3] = gA[3];

    const float4* gN = (const float4*)(Xb + (size_t)(n0 + row) * D_ + kb + col);
    float4* sN = (float4*)(&Ns[row * 36 + col]);
    sN[0] = gN[0]; sN[1] = gN[1]; sN[2] = gN[2]; sN[3] = gN[3];
    __syncthreads();

#pragma unroll
    for (int kk = 0; kk < 32; kk += 4) {
      v2f a[2], b[4];
#pragma unroll
      for (int t = 0; t < 2; t++) {
        const int mr = (wm * 32 + t * 16 + lm) * 36 + kk + 2 * h;
        a[t][0] = As[mr]; a[t][1] = As[mr + 1];
      }
#pragma unroll
      for (int t = 0; t < 4; t++) {
        const int nr = (wn * 64 + t * 16 + lm) * 36 + kk + 2 * h;
        b[t][0] = Ns[nr]; b[t][1] = Ns[nr + 1];
      }
#pragma unroll
      for (int i = 0; i < 2; i++)
#pragma unroll
        for (int j = 0; j < 4; j++) acc[i][j] = wmma4(a[i], b[j], acc[i][j]);
    }
    __syncthreads();
  }

  const float NEG_INF = -__builtin_inff();
#pragma unroll
  for (int i = 0; i < 2; i++)
#pragma unroll
    for (int j = 0; j < 4; j++) {
      const int jq = n0 + wn * 64 + j * 16 + lm;  // key index
      const bool pad = mask[bz * L_ + jq] != 0;
#pragma unroll
      for (int r = 0; r < 8; r++) {
        const int iq = m0 + wm * 32 + i * 16 + r + 8 * h;  // query index
        float v = acc[i][j][r];
        if (jq == iq) v = 0.0f;       // diag=False: zero diagonal first
        if (pad) v = NEG_INF;         // then mask padded keys
        S[(size_t)bz * L_ * L_ + (size_t)iq * L_ + jq] = v;
      }
    }
}

// ---------------------------------------------------------------------------
// Kernel 3: row softmax over L=2048 keys.  One 256-thread block per row.
// ---------------------------------------------------------------------------
__global__ __launch_bounds__(256)
void softmax_kernel(float* __restrict__ S) {
  float* p = S + (size_t)blockIdx.x * L_;
  const int tid = threadIdx.x;
  const int lane = tid & 31, wid = tid >> 5;
  __shared__ float redm[8];
  __shared__ float reds[8];

  float v[8];
  float m = -__builtin_inff();
#pragma unroll
  for (int k = 0; k < 8; k++) { v[k] = p[tid + 256 * k]; m = fmaxf(m, v[k]); }
#pragma unroll
  for (int off = 16; off > 0; off >>= 1) m = fmaxf(m, __shfl_xor(m, off, 32));
  if (lane == 0) redm[wid] = m;
  __syncthreads();
  float rowmax = redm[0];
#pragma unroll
  for (int w = 1; w < 8; w++) rowmax = fmaxf(rowmax, redm[w]);

  float s = 0.0f;
#pragma unroll
  for (int k = 0; k < 8; k++) { v[k] = __expf(v[k] - rowmax); s += v[k]; }
#pragma unroll
  for (int off = 16; off > 0; off >>= 1) s += __shfl_xor(s, off, 32);
  if (lane == 0) reds[wid] = s;
  __syncthreads();
  float tot = reds[0];
#pragma unroll
  for (int w = 1; w < 8; w++) tot += reds[w];
  const float inv = 1.0f / tot;
#pragma unroll
  for (int k = 0; k < 8; k++) p[tid + 256 * k] = v[k] * inv;
}

// ---------------------------------------------------------------------------
// Kernel 4: out[b][i][d] = sum_j alpha[b][i][j] * x[b][j][d]
//   Per batch GEMM: M=L, N=D, K=L.  Grid (L/128, D/128, B).
// ---------------------------------------------------------------------------
__global__ __launch_bounds__(256)
void av_kernel(const float* __restrict__ S, const float* __restrict__ X,
               float* __restrict__ OUT) {
  __shared__ float As[128 * 36];   // alpha tile [m][k]
  __shared__ float Bs[32 * 132];   // x tile [k][n], padded stride 132

  const int tid  = threadIdx.x;
  const int lane = tid & 31, wid = tid >> 5;
  const int wm = wid & 3, wn = wid >> 2;
  const int h = lane >> 4, lm = lane & 15;
  const int m0 = blockIdx.x * 128;
  const int n0 = blockIdx.y * 128;
  const int bz = blockIdx.z;
  const float* Ab = S + (size_t)bz * L_ * L_;
  const float* Xb = X + (size_t)bz * L_ * D_;

  const int arow = tid >> 1, acol = (tid & 1) * 16;   // 128x32 alpha tile
  const int brow = tid >> 3, bcol = (tid & 7) * 16;   // 32x128 x tile

  v8f acc[2][4];
  v8f z = {};
#pragma unroll
  for (int i = 0; i < 2; i++)
#pragma unroll
    for (int j = 0; j < 4; j++) acc[i][j] = z;

  for (int kb = 0; kb < L_; kb += 32) {
    const float4* gA = (const float4*)(Ab + (size_t)(m0 + arow) * L_ + kb + acol);
    float4* sA = (float4*)(&As[arow * 36 + acol]);
    sA[0] = gA[0]; sA[1] = gA[1]; sA[2] = gA[2]; sA[3] = gA[3];

    const float4* gB = (const float4*)(Xb + (size_t)(kb + brow) * D_ + n0 + bcol);
    float4* sB = (float4*)(&Bs[brow * 132 + bcol]);
    sB[0] = gB[0]; sB[1] = gB[1]; sB[2] = gB[2]; sB[3] = gB[3];
    __syncthreads();

#pragma unroll
    for (int kk = 0; kk < 32; kk += 4) {
      v2f a[2], b[4];
#pragma unroll
      for (int t = 0; t < 2; t++) {
        const int mr = (wm * 32 + t * 16 + lm) * 36 + kk + 2 * h;
        a[t][0] = As[mr]; a[t][1] = As[mr + 1];
      }
#pragma unroll
      for (int t = 0; t < 4; t++) {
        const int nr = (kk + 2 * h) * 132 + wn * 64 + t * 16 + lm;
        b[t][0] = Bs[nr]; b[t][1] = Bs[nr + 132];
      }
#pragma unroll
      for (int i = 0; i < 2; i++)
#pragma unroll
        for (int j = 0; j < 4; j++) acc[i][j] = wmma4(a[i], b[j], acc[i][j]);
    }
    __syncthreads();
  }

#pragma unroll
  for (int i = 0; i < 2; i++)
#pragma unroll
    for (int j = 0; j < 4; j++) {
      const int n = n0 + wn * 64 + j * 16 + lm;
#pragma unroll
      for (int r = 0; r < 8; r++) {
        const int m = m0 + wm * 32 + i * 16 + r + 8 * h;
        OUT[(size_t)bz * L_ * D_ + (size_t)m * D_ + n] = acc[i][j][r];
      }
    }
}

extern "C" void kernel_launch(void* const* d_in, const int* in_sizes, int n_in,
                              void* d_out, int out_size, void* d_ws, size_t ws_size,
                              hipStream_t stream) {
  const float*         x    = (const float*)d_in[0];          // [B,L,D] f32
  const unsigned char* mask = (const unsigned char*)d_in[1];  // [B,L] bool (1 byte)
  const float*         W    = (const float*)d_in[2];          // [D,D] f32
  const float*         bias = (const float*)d_in[3];          // [D] f32
  float* out = (float*)d_out;                                 // [B,L,D] f32

  float* xp = (float*)d_ws;                       // 64 MB
  float* S  = xp + (size_t)B_ * L_ * D_;          // 128 MB

  dim3 blk(256);
  proj_kernel<<<dim3((B_ * L_) / 128, D_ / 128), blk, 0, stream>>>(x, W, bias, xp);
  scores_kernel<<<dim3(L_ / 128, L_ / 128, B_), blk, 0, stream>>>(xp, mask, S);
  softmax_kernel<<<dim3(B_ * L_), blk, 0, stream>>>(S);
  av_kernel<<<dim3(L_ / 128, D_ / 128, B_), blk, 0, stream>>>(S, x, out);
}